// DynamicMaxPool_47253230191010
// MI455X (gfx1250) — compile-verified
//
#include <hip/hip_runtime.h>

// Problem constants (match reference: B=16, T=4096, C=256)
#define Bc 16
#define Tc 4096
#define Cc 256
#define TILE_C 16
#define NTILES (Cc / TILE_C)            // 16 column-tiles per batch
#define ROWP 17                         // LDS row pitch in dwords (16 data + 1 TDM pad) -> bank-conflict-free
#define NTHREADS 256
#define NWAVES (NTHREADS / 32)          // 8 wave32s
#define COLS_PER_WAVE (TILE_C / NWAVES) // 2

typedef unsigned int u32x4 __attribute__((ext_vector_type(4)));
typedef int          i32x4 __attribute__((ext_vector_type(4)));
typedef int          i32x8 __attribute__((ext_vector_type(8)));

// Hard requirement, but only for the DEVICE pass: the host x86 pass of hipcc
// does not declare AMDGPU builtins, so __has_builtin is false there by design.
#if defined(__HIP_DEVICE_COMPILE__)
#  if !__has_builtin(__builtin_amdgcn_tensor_load_to_lds)
#    error "gfx1250 device pass lacks __builtin_amdgcn_tensor_load_to_lds"
#  endif
#  if !__has_builtin(__builtin_amdgcn_s_wait_tensorcnt)
#    error "gfx1250 device pass lacks __builtin_amdgcn_s_wait_tensorcnt"
#  endif
#endif

// Map float bits to a totally-ordered unsigned key (ascending key == ascending float).
__device__ __forceinline__ unsigned okey(unsigned u) {
  return (u & 0x80000000u) ? ~u : (u | 0x80000000u);
}

__global__ void __launch_bounds__(NTHREADS)
kmax_pool_kernel(const float* __restrict__ x,
                 const int* __restrict__ lengths,
                 const int* __restrict__ pool_ranges,
                 const int* __restrict__ top_k_p,
                 const int* __restrict__ layer_p,
                 const int* __restrict__ total_layers_p,
                 float* __restrict__ out) {
  extern __shared__ unsigned smem[];
  unsigned* tile = smem;                 // Tc * ROWP dwords (padded tile)
  unsigned* hist = smem + Tc * ROWP;     // NWAVES * 256 dwords

  const int tid  = threadIdx.x;
  const int lane = tid & 31;
  const int wid  = tid >> 5;
  const int b    = blockIdx.x / NTILES;
  const int c0   = (blockIdx.x % NTILES) * TILE_C;

  const int R  = pool_ranges[b];
  const int L  = lengths[b];
  const int tk = top_k_p[0];
  const int ly = layer_p[0];
  const int tl = total_layers_p[0];
  // 32-bit ceil-div (num*L <= ~12K): avoids the i64 software-division blob in SALU.
  const int num = tl - ly;
  int k = (num * L + tl - 1) / tl;
  if (k < tk) k = tk;
  if (k > R)  k = R;                     // survivors per column; uniform within batch

  // ---------------- Stage tile [Tc rows x TILE_C cols] global -> LDS via TDM ----------------
  const unsigned long long gaddr =
      (unsigned long long)(const void*)(x + ((size_t)b * Tc * Cc + c0));

#if __has_builtin(__builtin_amdgcn_tensor_load_to_lds)
  if (wid == 0) {
    const unsigned lds_base = (unsigned)(unsigned long long)(const void*)tile;
    // D# group0: count=1 (user), lds_addr, global_addr[56:0], type=2 ("image")
    u32x4 g0 = { 1u,
                 lds_base,
                 (unsigned)(gaddr & 0xffffffffull),
                 (unsigned)((gaddr >> 32) & 0x1ffffffull) | (2u << 30) };
    // D# group1: data_size=4B(code 2), pad_enable=1, pad_interval=16dw(code 3), pad_amount=1dw(code 0)
    //   tensor_dim0=Cc, tensor_dim1=Tc, tile_dim0=TILE_C, tile_dim1=Tc, tensor_dim0_stride=Cc
    i32x8 g1 = { (int)((2u << 16) | (1u << 20) | (3u << 22)),
                 (int)((unsigned)Cc << 16),      // tensor_dim0[15:0] at bits[63:48]
                 (int)((unsigned)Tc << 16),      // dim0 hi = 0 | tensor_dim1[15:0]
                 (int)((unsigned)TILE_C << 16),  // dim1 hi = 0 | tile_dim0
                 (int)Tc,                        // tile_dim1 | tile_dim2=0
                 (int)Cc,                        // tensor_dim0_stride[31:0]
                 0, 0 };                         // stride0 hi | stride1 (2D: unused)
    i32x4 gz4 = { 0, 0, 0, 0 };
#if __clang_major__ >= 23
    i32x8 gz8 = { 0, 0, 0, 0, 0, 0, 0, 0 };
    __builtin_amdgcn_tensor_load_to_lds(g0, g1, gz4, gz4, gz8, 0);
#else
    __builtin_amdgcn_tensor_load_to_lds(g0, g1, gz4, gz4, 0);
#endif
    __builtin_amdgcn_s_wait_tensorcnt(0);
  }
#else
  // Host-parse-only fallback (device pass always takes the TDM branch above).
  {
    const unsigned* src = (const unsigned*)(x + ((size_t)b * Tc * Cc + c0));
    for (int i = tid; i < Tc * TILE_C; i += NTHREADS) {
      const int r = i >> 4, c = i & 15;
      tile[r * ROWP + c] = src[(size_t)r * Cc + c];
    }
  }
#endif
  __syncthreads();

  // ---------------- Per-column exact radix select + stable compaction ----------------
  unsigned* h = hist + wid * 256;
  const unsigned ltmask = (1u << lane) - 1u;

  for (int cc = 0; cc < COLS_PER_WAVE; ++cc) {
    const int ch = wid * COLS_PER_WAVE + cc;   // column within tile, owned by this wave

    unsigned prefix = 0, pmask = 0;
    int remaining = k;
    for (int round = 0; round < 4; ++round) {
      const int shift = 24 - 8 * round;
      for (int i = lane; i < 256; i += 32) h[i] = 0u;   // per-wave LDS ops are in-order
      for (int t = lane; t < Tc; t += 32) {
        if (t < R) {
          const unsigned ky = okey(tile[t * ROWP + ch]);
          if ((ky & pmask) == prefix)
            atomicAdd(&h[(ky >> shift) & 0xFFu], 1u);
        }
      }
      // descending cumulative scan (all lanes redundantly -> uniform result)
      unsigned cum = 0; int sel = 0;
      for (int bin = 255; bin >= 0; --bin) {
        const unsigned cnt = h[bin];
        if (cum + cnt >= (unsigned)remaining) { sel = bin; break; }
        cum += cnt;
      }
      remaining -= (int)cum;
      prefix |= (unsigned)sel << shift;
      pmask  |= 0xFFu << shift;
    }
    const unsigned thr = prefix;   // exact k-th largest key
    const int quota = remaining;   // how many threshold-equal entries survive (earliest rows)

    // Stable in-place compaction: survivors -> rows [0, k) of this column.
    int pos = 0, eqc = 0;
    for (int t0 = 0; t0 < Tc; t0 += 32) {
      const int t = t0 + lane;
      const unsigned u  = tile[t * ROWP + ch];   // all lanes load before any lane stores
      const unsigned ky = okey(u);
      const bool valid = (t < R);
      const bool gt  = valid && (ky > thr);
      const bool eqf = valid && (ky == thr);
      const unsigned em = __builtin_amdgcn_ballot_w32(eqf);
      const int myeq = eqc + __builtin_popcount(em & ltmask);
      const bool keep = gt || (eqf && (myeq < quota));
      const unsigned km = __builtin_amdgcn_ballot_w32(keep);
      const int mypos = pos + __builtin_popcount(km & ltmask);
      if (keep) tile[mypos * ROWP + ch] = u;     // mypos <= t always -> in-place safe
      pos += __builtin_popcount(km);
      eqc += __builtin_popcount(em);
    }
  }
  __syncthreads();

  // ---------------- Coalesced output: rows [0,k) compacted, rows [k,T) zero ----------------
  const size_t obase = (size_t)b * Tc * Cc + c0;
  for (int i = tid; i < k * TILE_C; i += NTHREADS) {
    const int r = i >> 4, c = i & 15;
    out[obase + (size_t)r * Cc + c] = __uint_as_float(tile[r * ROWP + c]);
  }
  for (int i = tid; i < (Tc - k) * TILE_C; i += NTHREADS) {
    const int r = k + (i >> 4), c = i & 15;
    out[obase + (size_t)r * Cc + c] = 0.0f;
  }
  if (tid == 0 && (blockIdx.x % NTILES) == 0) {
    out[(size_t)Bc * Tc * Cc + b] = (float)k;   // pool_result_ranges
  }
}

extern "C" void kernel_launch(void* const* d_in, const int* in_sizes, int n_in,
                              void* d_out, int out_size, void* d_ws, size_t ws_size,
                              hipStream_t stream) {
  (void)in_sizes; (void)n_in; (void)d_ws; (void)ws_size; (void)out_size;
  const float* x            = (const float*)d_in[0];
  const int*   lengths      = (const int*)d_in[1];
  const int*   pool_ranges  = (const int*)d_in[2];
  const int*   top_k        = (const int*)d_in[3];
  const int*   layer        = (const int*)d_in[4];
  const int*   total_layers = (const int*)d_in[5];
  float* out = (float*)d_out;

  const size_t shmem = (size_t)(Tc * ROWP + NWAVES * 256) * sizeof(unsigned); // ~280 KiB (<= 320 KB WGP LDS)
  (void)hipFuncSetAttribute((const void*)kmax_pool_kernel,
                            hipFuncAttributeMaxDynamicSharedMemorySize, (int)shmem);

  hipLaunchKernelGGL(kmax_pool_kernel,
                     dim3(Bc * NTILES), dim3(NTHREADS), shmem, stream,
                     x, lengths, pool_ranges, top_k, layer, total_layers, out);
}